// OptimizedDifferentiableMutualInformationLoss_38293928411580
// MI455X (gfx1250) — compile-verified
//
#include <hip/hip_runtime.h>

typedef __attribute__((ext_vector_type(2))) float v2f;
typedef __attribute__((ext_vector_type(8))) float v8f;

#define TILE 16
#define JCHUNKS 8
#define WAVES_PER_BLOCK 8
#define LOG2E 1.44269504088896340736f
#define EPSV 1e-8f

// ---------------------------------------------------------------------------
// Kernel 0: zero the row-sum workspace (harness poisons d_ws with 0xAA).
// ---------------------------------------------------------------------------
__global__ void mi_zero_ws(float* __restrict__ ws, int n3) {
    int i = blockIdx.x * blockDim.x + threadIdx.x;
    if (i < n3) ws[i] = 0.0f;
}

// ---------------------------------------------------------------------------
// Kernel 1: WMMA-tiled pairwise Gaussian-kernel row sums.
// One wave owns one 16-row i-tile and one chunk of j-tiles.
// A-row (y):  [100L*yi, -50L*yi^2, -50L, 0]
// B-col (y):  [yj, 1, yj^2, 0]
// => WMMA(A,B)[i][j] = -50*L*(yi-yj)^2  (the exp2 exponent, ready to use)
// k_joint = k_y * k_p (since squared distances add in the joint space).
// ---------------------------------------------------------------------------
__global__ __launch_bounds__(256) void mi_kde_rowsums(
    const float* __restrict__ p,   // predicted_y
    const float* __restrict__ y,   // y_observed
    float* __restrict__ wsY, float* __restrict__ wsP, float* __restrict__ wsJ,
    int tiles_per_chunk)
{
    const int lane  = threadIdx.x & 31;
    const int wave  = threadIdx.x >> 5;
    const int gid   = blockIdx.x * WAVES_PER_BLOCK + wave;
    const int itile = gid >> 3;               // / JCHUNKS
    const int jchnk = gid & (JCHUNKS - 1);
    const int m     = lane & 15;
    const bool hi   = (lane >= 16);

    const int i0 = itile * TILE;
    const float yi = y[i0 + m];
    const float pi = p[i0 + m];
    const float cY = 100.0f * LOG2E;          // linear-term scale
    const float cS = -50.0f * LOG2E;          // square-term scale

    // A operands (16x4 f32 layout: lanes 0-15 -> K=0,1 ; lanes 16-31 -> K=2,3)
    v2f Ay, Ap;
    Ay.x = hi ? cS   : cY * yi;
    Ay.y = hi ? 0.0f : cS * yi * yi;
    Ap.x = hi ? cS   : cY * pi;
    Ap.y = hi ? 0.0f : cS * pi * pi;

    v8f accY = {}; v8f accP = {}; v8f accJ = {};

    int j0 = jchnk * tiles_per_chunk * TILE;
    for (int t = 0; t < tiles_per_chunk; ++t, j0 += TILE) {
        const float yj = y[j0 + m];
        const float pj = p[j0 + m];
        // B operands (4x16 f32 layout: lanes 0-15 -> K=0,1 ; lanes 16-31 -> K=2,3)
        v2f By, Bp;
        By.x = hi ? yj * yj : yj;
        By.y = hi ? 0.0f    : 1.0f;
        Bp.x = hi ? pj * pj : pj;
        Bp.y = hi ? 0.0f    : 1.0f;

        v8f zero = {};
        v8f ty = __builtin_amdgcn_wmma_f32_16x16x4_f32(
            false, Ay, false, By, (short)0, zero, false, false);
        v8f tp = __builtin_amdgcn_wmma_f32_16x16x4_f32(
            false, Ap, false, Bp, (short)0, zero, false, false);

        #pragma unroll
        for (int r = 0; r < 8; ++r) {
            // clamp d2 >= 0  <=>  exponent <= 0
            float ey = __builtin_amdgcn_exp2f(fminf(ty[r], 0.0f));
            float ep = __builtin_amdgcn_exp2f(fminf(tp[r], 0.0f));
            accY[r] += ey;
            accP[r] += ep;
            accJ[r] += ey * ep;   // joint kernel = product of marginals
        }
    }

    // Reduce over the j (N) dimension: sum across each 16-lane half.
    #pragma unroll
    for (int r = 0; r < 8; ++r) {
        float sy = accY[r], sp = accP[r], sj = accJ[r];
        #pragma unroll
        for (int mask = 1; mask < 16; mask <<= 1) {
            sy += __shfl_xor(sy, mask, 32);
            sp += __shfl_xor(sp, mask, 32);
            sj += __shfl_xor(sj, mask, 32);
        }
        if (m == 0) {
            const int row = i0 + r + (hi ? 8 : 0);  // C/D layout: VGPR r -> rows r, r+8
            atomicAdd(&wsY[row], sy);
            atomicAdd(&wsP[row], sp);
            atomicAdd(&wsJ[row], sj);
        }
    }
}

// ---------------------------------------------------------------------------
// Kernel 2: logs + scalar.  out = h_joint - h_y - h_p
//         = (1/N) * sum_i [ log(ksumY_i) + log(ksumP_i) - log(ksumJ_i) ]
// ---------------------------------------------------------------------------
__global__ __launch_bounds__(256) void mi_finalize(
    const float* __restrict__ wsY, const float* __restrict__ wsP,
    const float* __restrict__ wsJ, float* __restrict__ out, int n)
{
    __shared__ float red[256];
    const float inv = 1.0f / (float)(n - 1);
    float local = 0.0f;
    for (int r = threadIdx.x; r < n; r += 256) {
        float sy = (wsY[r] - 1.0f) * inv;   // remove self-kernel, average
        float sp = (wsP[r] - 1.0f) * inv;
        float sj = (wsJ[r] - 1.0f) * inv;
        local += logf(sy + EPSV) + logf(sp + EPSV) - logf(sj + EPSV);
    }
    red[threadIdx.x] = local;
    __syncthreads();
    for (int s = 128; s > 0; s >>= 1) {
        if (threadIdx.x < s) red[threadIdx.x] += red[threadIdx.x + s];
        __syncthreads();
    }
    if (threadIdx.x == 0) out[0] = red[0] / (float)n;
}

// ---------------------------------------------------------------------------
extern "C" void kernel_launch(void* const* d_in, const int* in_sizes, int n_in,
                              void* d_out, int out_size, void* d_ws, size_t ws_size,
                              hipStream_t stream) {
    const float* pred = (const float*)d_in[0];   // predicted_y
    const float* yobs = (const float*)d_in[1];   // y_observed
    float* out = (float*)d_out;

    const int n = in_sizes[0];                   // 8192
    const int ntiles = n / TILE;                 // 512
    const int tiles_per_chunk = ntiles / JCHUNKS;// 64
    const int total_waves = ntiles * JCHUNKS;    // 4096
    const int blocks = total_waves / WAVES_PER_BLOCK; // 512

    float* wsY = (float*)d_ws;
    float* wsP = wsY + n;
    float* wsJ = wsP + n;

    mi_zero_ws<<<(3 * n + 255) / 256, 256, 0, stream>>>(wsY, 3 * n);
    mi_kde_rowsums<<<blocks, 256, 0, stream>>>(pred, yobs, wsY, wsP, wsJ,
                                               tiles_per_chunk);
    mi_finalize<<<1, 256, 0, stream>>>(wsY, wsP, wsJ, out, n);
}